// GraphAttentionLayer_65171833749775
// MI455X (gfx1250) — compile-verified
//
#include <hip/hip_runtime.h>
#include <hip/hip_bf16.h>

#define N_NODES 8192
#define DIN 256
#define DOUT 128
#define NEG_SLOPE 0.01f
#define MASK_VAL (-9.0e15f)

typedef float v2f __attribute__((ext_vector_type(2)));
typedef float v8f __attribute__((ext_vector_type(8)));

// leaky_relu then mask; exp(MASK_VAL) underflows to exactly 0, matching the
// reference's masked-softmax semantics without a row-max pass.
__device__ __forceinline__ float lrelu_mask(float e, int a) {
    e = (e > 0.f) ? e : NEG_SLOPE * e;
    return (a > 0) ? e : MASK_VAL;
}

// ---------------------------------------------------------------------------
// Kernel 1: h_T = (x @ W + bW)^T   stored as [DOUT][N] for contiguous-K reads
// One wave per 16x16 output tile via v_wmma_f32_16x16x4_f32.
// ---------------------------------------------------------------------------
__global__ __launch_bounds__(256) void gat_h_gemm(
    const float* __restrict__ x, const float* __restrict__ W,
    const float* __restrict__ bW, float* __restrict__ hT)
{
    const int lane = threadIdx.x & 31;
    const int wave = threadIdx.x >> 5;
    const int tile = blockIdx.x * 8 + wave;        // 4096 tiles total
    const int row0 = (tile >> 3) << 4;
    const int col0 = (tile & 7) << 4;

    const int half = lane >> 4;
    const int lr   = lane & 15;
    const int m    = row0 + lr;                    // A-row for this lane
    const int n    = col0 + lr;                    // B-col for this lane

    v8f acc = {};
    for (int k0 = 0; k0 < DIN; k0 += 8) {
        const int kb = k0 + (half << 2);           // this lane-half's 4 ks
        const float4 a4 = *(const float4*)(x + (size_t)m * DIN + kb);
        float4 b4;
        b4.x = W[(size_t)(kb + 0) * DOUT + n];
        b4.y = W[(size_t)(kb + 1) * DOUT + n];
        b4.z = W[(size_t)(kb + 2) * DOUT + n];
        b4.w = W[(size_t)(kb + 3) * DOUT + n];
        v2f A0 = {a4.x, a4.y}, B0 = {b4.x, b4.y};
        v2f A1 = {a4.z, a4.w}, B1 = {b4.z, b4.w};
        acc = __builtin_amdgcn_wmma_f32_16x16x4_f32(
            false, A0, false, B0, (short)0, acc, false, false);
        acc = __builtin_amdgcn_wmma_f32_16x16x4_f32(
            false, A1, false, B1, (short)0, acc, false, false);
    }
    const float bias = bW[n];
    // rows row0+8*half .. +7 are contiguous along h_T's fast axis
    float4 lo, hi;
    lo.x = acc[0] + bias; lo.y = acc[1] + bias;
    lo.z = acc[2] + bias; lo.w = acc[3] + bias;
    hi.x = acc[4] + bias; hi.y = acc[5] + bias;
    hi.z = acc[6] + bias; hi.w = acc[7] + bias;
    float* dst = hT + (size_t)n * N_NODES + row0 + (half << 3);
    *(float4*)(dst)     = lo;
    *(float4*)(dst + 4) = hi;
}

// ---------------------------------------------------------------------------
// Kernel 2: s_src[i] = h[i,:]·a1 ; s_dst[i] = h[i,:]·a2 (reads h_T, coalesced)
// ---------------------------------------------------------------------------
__global__ __launch_bounds__(256) void gat_scores(
    const float* __restrict__ hT, const float* __restrict__ a1,
    const float* __restrict__ a2, float* __restrict__ s_src,
    float* __restrict__ s_dst)
{
    const int i = blockIdx.x * blockDim.x + threadIdx.x;
    if (i >= N_NODES) return;
    float s1 = 0.f, s2 = 0.f;
#pragma unroll 4
    for (int d = 0; d < DOUT; ++d) {
        const float v = hT[(size_t)d * N_NODES + i];   // coalesced across i
        s1 = fmaf(v, a1[d], s1);
        s2 = fmaf(v, a2[d], s2);
    }
    s_src[i] = s1;
    s_dst[i] = s2;
}

// ---------------------------------------------------------------------------
// Kernel 3 (fused attention): out = softmax(e) @ h in a SINGLE adj pass.
//   p_ij = exp(lrelu(s_src+s_dst+ba) masked)  -- no row-max needed (bounded e;
//          masked entries underflow to exactly 0)
//   row normalizer computed for free by an extra WMMA against a ones-B tile:
//          acc_sum[v] = sum_j p[row,j] (replicated over columns)
// One wave per 16-row block; 8 v8f accs cover DOUT=128 + 1 v8f for row sums.
// K chunk of 8: lane-half owns contiguous js -> adj/s_dst are one b128 each,
// each tile's B operand is ONE float4 from h_T feeding two WMMAs.
// ---------------------------------------------------------------------------
__global__ __launch_bounds__(256) void gat_attn_gemm(
    const int* __restrict__ adj, const float* __restrict__ hT,
    const float* __restrict__ s_src, const float* __restrict__ s_dst,
    const float* __restrict__ ba_p, float* __restrict__ out)
{
    const int lane = threadIdx.x & 31;
    const int wave = threadIdx.x >> 5;
    const int row0 = (blockIdx.x * 8 + wave) << 4;

    const int half = lane >> 4;
    const int lr   = lane & 15;
    const int ri   = row0 + lr;                    // A-row of this lane
    const float ba = ba_p[0];
    const float si = s_src[ri] + ba;
    const size_t arow = (size_t)ri * N_NODES;
    const float* hbase = hT + (size_t)lr * N_NODES;  // + t*16*N_NODES per tile

    v8f acc[8];
#pragma unroll
    for (int t = 0; t < 8; ++t) acc[t] = (v8f){};
    v8f accs = {};                                  // row-sum accumulator
    const v2f Bones = {1.f, 1.f};

    for (int j0 = 0; j0 < N_NODES; j0 += 8) {
        const int kb = j0 + (half << 2);           // this lane-half's 4 js
        __builtin_prefetch(adj + arow + kb + 512, 0, 1);  // global_prefetch_b8

        const int4   aj = *(const int4*)(adj + arow + kb);
        const float4 sd = *(const float4*)(s_dst + kb);

        const float p0 = __expf(lrelu_mask(si + sd.x, aj.x));
        const float p1 = __expf(lrelu_mask(si + sd.y, aj.y));
        const float p2 = __expf(lrelu_mask(si + sd.z, aj.z));
        const float p3 = __expf(lrelu_mask(si + sd.w, aj.w));
        const v2f A0 = {p0, p1};
        const v2f A1 = {p2, p3};

        // free row-normalizer: D[m][*] += sum of this chunk's p[m][:]
        accs = __builtin_amdgcn_wmma_f32_16x16x4_f32(
            false, A0, false, Bones, (short)0, accs, false, false);
        accs = __builtin_amdgcn_wmma_f32_16x16x4_f32(
            false, A1, false, Bones, (short)0, accs, false, false);

#pragma unroll
        for (int t = 0; t < 8; ++t) {
            const float4 hb =
                *(const float4*)(hbase + (size_t)(t << 4) * N_NODES + kb);
            const v2f B0 = {hb.x, hb.y};
            const v2f B1 = {hb.z, hb.w};
            acc[t] = __builtin_amdgcn_wmma_f32_16x16x4_f32(
                false, A0, false, B0, (short)0, acc[t], false, false);
            acc[t] = __builtin_amdgcn_wmma_f32_16x16x4_f32(
                false, A1, false, B1, (short)0, acc[t], false, false);
        }
    }

#pragma unroll
    for (int v = 0; v < 8; ++v) {
        const int row = row0 + v + (half << 3);
        const float s = accs[v];
        const float inv = (s > 0.f) ? 1.0f / s : 0.f;
#pragma unroll
        for (int t = 0; t < 8; ++t) {
            out[(size_t)row * DOUT + (t << 4) + lr] = acc[t][v] * inv;
        }
    }
}

// ---------------------------------------------------------------------------
extern "C" void kernel_launch(void* const* d_in, const int* in_sizes, int n_in,
                              void* d_out, int out_size, void* d_ws, size_t ws_size,
                              hipStream_t stream)
{
    const float* x   = (const float*)d_in[0];
    const int*   adj = (const int*)  d_in[1];
    const float* W   = (const float*)d_in[2];
    const float* bW  = (const float*)d_in[3];
    const float* a1  = (const float*)d_in[4];
    const float* a2  = (const float*)d_in[5];
    const float* ba  = (const float*)d_in[6];
    float* out = (float*)d_out;

    float* ws    = (float*)d_ws;
    float* hT    = ws;                                   // 128*8192 floats (4 MB)
    float* s_src = hT + (size_t)DOUT * N_NODES;          // 8192
    float* s_dst = s_src + N_NODES;                      // 8192

    // K1: 4096 tiles, 8 waves/block -> 512 blocks of 256
    gat_h_gemm<<<512, 256, 0, stream>>>(x, W, bW, hT);
    // K2: one thread per row
    gat_scores<<<N_NODES / 256, 256, 0, stream>>>(hT, a1, a2, s_src, s_dst);
    // K3: 512 row-blocks of 16 rows, 8 waves/block -> 64 blocks, single adj pass
    gat_attn_gemm<<<(N_NODES / 16) / 8, 256, 0, stream>>>(adj, hT, s_src, s_dst, ba, out);
}